// GraphFilterDown_45775761440991
// MI455X (gfx1250) — compile-verified
//
#include <hip/hip_runtime.h>

// ---------------------------------------------------------------------------
// GraphFilterDown on gfx1250 (MI455X):
//   z1 = x@S ; z2 = z1@S ; z3 = z2@S   (three 4096^3 fp32 GEMMs, WMMA f32)
//   y  = 0.5 * sum_k  group4-reduce_f( w[k,f] * z_k )   (z0 = x)
//
// GEMM pipeline: async global->LDS copies (ASYNCcnt) with per-lane LDS
// scatter (B transposed in-flight), LDS double-buffering with the pipeline
// loop unrolled 2x (compile-time buffer indices -> no cndmask selects),
// one workgroup barrier per K-slab, v_wmma_f32_16x16x4_f32 compute fed by
// single ds_load_2addr_b64 fragment loads.
// ---------------------------------------------------------------------------

typedef __attribute__((ext_vector_type(2))) float v2f;
typedef __attribute__((ext_vector_type(8))) float v8f;

#define NN   4096   // matrix dimension (M = T*FIN = N = K = 4096)
#define BM   128
#define BN   128
#define BK   16
#define LDSA 20     // padded A stride in floats (80B: 16B aligned b128 stores)
#define LDSB 18     // padded B stride in floats (72B: 8B aligned k-pairs,
                    // 18n mod 64 -> 32 distinct banks for consecutive n)

// gfx1250 async global->LDS copies. VDST carries the per-lane LDS byte
// address (low 32 bits of the generic shared pointer), VADDR the 64-bit
// global address. Tracked by ASYNCcnt; per-lane LDS scatter is legal.
__device__ __forceinline__ void async_copy_b128(unsigned lds_addr, const float* gaddr) {
    asm volatile("global_load_async_to_lds_b128 %0, %1, off"
                 :: "v"(lds_addr), "v"((unsigned long long)(uintptr_t)gaddr)
                 : "memory");
}
__device__ __forceinline__ void async_copy_b32(unsigned lds_addr, const float* gaddr) {
    asm volatile("global_load_async_to_lds_b32 %0, %1, off"
                 :: "v"(lds_addr), "v"((unsigned long long)(uintptr_t)gaddr)
                 : "memory");
}
__device__ __forceinline__ void wait_async_zero() {
    asm volatile("s_wait_asynccnt 0x0" ::: "memory");
}

// ---------------------------------------------------------------------------
// Tiled fp32 GEMM  C[4096x4096] = A[4096x4096] * B[4096x4096]
// 256 threads = 8 waves (wave32); each wave owns a 32x64 sub-tile
// (2x4 grid of 16x16 WMMA tiles, 8 independent v8f accumulators).
// ---------------------------------------------------------------------------
__global__ __launch_bounds__(256)
void gemm4096_wmma_f32(const float* __restrict__ A,
                       const float* __restrict__ B,
                       float* __restrict__ C) {
    __shared__ __align__(16) float As[2][BM * LDSA];  // A: [m][k], stride 20
    __shared__ __align__(16) float Bs[2][BN * LDSB];  // B: [n][k] (transposed), stride 18

    const int tid   = threadIdx.x;
    const int lane  = tid & 31;
    const int wave  = tid >> 5;
    const int waveM = wave >> 1;        // 0..3 -> 32-row band
    const int waveN = wave & 1;         // 0..1 -> 64-col band

    const int rowBase = blockIdx.y * BM;
    const int colBase = blockIdx.x * BN;

    const int lmod  = lane & 15;        // 0..15
    const int lhalf = lane >> 4;        // 0 or 1

    v8f acc[2][4];
#pragma unroll
    for (int mi = 0; mi < 2; ++mi)
#pragma unroll
        for (int ni = 0; ni < 4; ++ni)
            acc[mi][ni] = (v8f){0.f,0.f,0.f,0.f,0.f,0.f,0.f,0.f};

    // --- A async slots: 128x16 floats = 512 x 16B chunks; wave owns 64 -----
    const int cA0 = wave * 64 + lane;
    const int cA1 = cA0 + 32;
    const int amA0 = cA0 >> 2, akA0 = (cA0 & 3) * 4;
    const int amA1 = cA1 >> 2, akA1 = (cA1 & 3) * 4;
    const float* gA0 = A + (size_t)(rowBase + amA0) * NN + akA0;
    const float* gA1 = A + (size_t)(rowBase + amA1) * NN + akA1;
    unsigned ldsA0[2], ldsA1[2];
#pragma unroll
    for (int b = 0; b < 2; ++b) {
        ldsA0[b] = (unsigned)(uintptr_t)&As[b][amA0 * LDSA + akA0];
        ldsA1[b] = (unsigned)(uintptr_t)&As[b][amA1 * LDSA + akA1];
    }

    // --- B async slots: transpose-in-flight via per-lane b32 scatter -------
    // wave owns k-rows {2*wave, 2*wave+1}; element (j,q): k=2w+j, n=q*32+lane
    const int bK = 2 * wave;
    const float* gB[2];
    gB[0] = B + (size_t)(bK + 0) * NN + colBase + lane;
    gB[1] = B + (size_t)(bK + 1) * NN + colBase + lane;
    unsigned ldsB[2][2][4];                              // [buf][j][q]
#pragma unroll
    for (int b = 0; b < 2; ++b)
#pragma unroll
        for (int j = 0; j < 2; ++j)
#pragma unroll
            for (int q = 0; q < 4; ++q)
                ldsB[b][j][q] =
                    (unsigned)(uintptr_t)&Bs[b][(q * 32 + lane) * LDSB + bK + j];

    // issue all 10 async copies for the slab starting at K-offset ka into buf b
    auto issue = [&](int b, size_t ka) {
        async_copy_b128(ldsA0[b], gA0 + ka);
        async_copy_b128(ldsA1[b], gA1 + ka);
#pragma unroll
        for (int j = 0; j < 2; ++j)
#pragma unroll
            for (int q = 0; q < 4; ++q)
                async_copy_b32(ldsB[b][j][q], gB[j] + ka * NN + q * 32);
    };

    // 4 WMMA k-substeps of K=4 on buffer (as, bs)
    auto compute = [&](const float* as, const float* bs) {
#pragma unroll
        for (int sub = 0; sub < 4; ++sub) {
            const int k0 = sub * 4;  // even -> b64-aligned frag reads

            // A frag (16x4 f32): lane l -> A[m=l%16][k0 + 2*(l/16) + {0,1}]
            v2f afrag[2];
#pragma unroll
            for (int mi = 0; mi < 2; ++mi) {
                const int m = waveM * 32 + mi * 16 + lmod;
                afrag[mi] = *(const v2f*)(&as[m * LDSA + k0 + 2 * lhalf]);
            }
            // B frag (4x16 f32): lane l -> B[k0 + 2*(l/16) + {0,1}][n=l%16]
            v2f bfrag[4];
#pragma unroll
            for (int ni = 0; ni < 4; ++ni) {
                const int n = waveN * 64 + ni * 16 + lmod;
                bfrag[ni] = *(const v2f*)(&bs[n * LDSB + k0 + 2 * lhalf]);
            }
#pragma unroll
            for (int mi = 0; mi < 2; ++mi)
#pragma unroll
                for (int ni = 0; ni < 4; ++ni)
                    acc[mi][ni] = __builtin_amdgcn_wmma_f32_16x16x4_f32(
                        false, afrag[mi], false, bfrag[ni],
                        (short)0, acc[mi][ni], false, false);
        }
    };

    // --- prologue: start slab 0 into buffer 0 ------------------------------
    issue(0, 0);

    // NN/BK = 256 slabs, processed in pairs with hard-coded buffer indices.
    for (int kb = 0; kb < NN; kb += 2 * BK) {
        // slab kb -> buffer 0
        wait_async_zero();          // own copies for this slab are in LDS
        __syncthreads();            // all copies visible; buf1 compute done
        issue(1, (size_t)(kb + BK));            // always in range (kb <= NN-32)
        compute(As[0], Bs[0]);

        // slab kb+BK -> buffer 1
        wait_async_zero();
        __syncthreads();
        if (kb + 2 * BK < NN)
            issue(0, (size_t)(kb + 2 * BK));
        compute(As[1], Bs[1]);
    }

    // ---- epilogue: D layout (ISA 7.12.2): row = v + 8*(lane/16), col = lane%16
#pragma unroll
    for (int mi = 0; mi < 2; ++mi) {
#pragma unroll
        for (int ni = 0; ni < 4; ++ni) {
            const int col  = colBase + waveN * 64 + ni * 16 + lmod;
            const int rowb = rowBase + waveM * 32 + mi * 16 + 8 * lhalf;
            v8f a = acc[mi][ni];
#pragma unroll
            for (int v = 0; v < 8; ++v)
                C[(size_t)(rowb + v) * NN + col] = a[v];
        }
    }
}

// ---------------------------------------------------------------------------
// y[t,fo,n] (+)= 0.5 * sum_{m=0..3} w[k, 4*fo+m] * src[t, 4*fo+m, n]
// src is [4096, 4096] (rows = t*64 + f). Vectorized float4 along n.
// beta==0: overwrite (initializes y); beta==1: accumulate.
// ---------------------------------------------------------------------------
__global__ __launch_bounds__(256)
void weighted_group_reduce(const float* __restrict__ src,
                           const float* __restrict__ wrow,  // 64 weights (row k)
                           float* __restrict__ y,
                           int beta) {
    const int gid = blockIdx.x * 256 + threadIdx.x;     // 0 .. 64*16*1024-1
    const int n4 = gid & 1023;          // N/4 = 1024
    const int fo = (gid >> 10) & 15;    // Fout = 16
    const int t  = gid >> 14;           // T = 64
    const int f0 = fo * 4;

    const float w0 = wrow[f0 + 0];
    const float w1 = wrow[f0 + 1];
    const float w2 = wrow[f0 + 2];
    const float w3 = wrow[f0 + 3];

    const size_t rbase = ((size_t)t * 64 + f0) * (NN / 4) + n4;  // float4 units
    const float4* s4 = (const float4*)src;
    float4 s0 = s4[rbase + 0 * (NN / 4)];
    float4 s1 = s4[rbase + 1 * (NN / 4)];
    float4 s2 = s4[rbase + 2 * (NN / 4)];
    float4 s3 = s4[rbase + 3 * (NN / 4)];

    float4 r;
    r.x = 0.5f * (w0 * s0.x + w1 * s1.x + w2 * s2.x + w3 * s3.x);
    r.y = 0.5f * (w0 * s0.y + w1 * s1.y + w2 * s2.y + w3 * s3.y);
    r.z = 0.5f * (w0 * s0.z + w1 * s1.z + w2 * s2.z + w3 * s3.z);
    r.w = 0.5f * (w0 * s0.w + w1 * s1.w + w2 * s2.w + w3 * s3.w);

    float4* y4 = (float4*)y;
    if (beta) {
        float4 p = y4[gid];
        r.x += p.x; r.y += p.y; r.z += p.z; r.w += p.w;
    }
    y4[gid] = r;
}

// ---------------------------------------------------------------------------
extern "C" void kernel_launch(void* const* d_in, const int* in_sizes, int n_in,
                              void* d_out, int out_size, void* d_ws, size_t ws_size,
                              hipStream_t stream) {
    const float* x = (const float*)d_in[0];   // [T, FIN, N]  = [4096, 4096] flat
    const float* S = (const float*)d_in[1];   // [N, N]
    const float* W = (const float*)d_in[2];   // [K, FIN] = [4, 64]
    float* y = (float*)d_out;                 // [T, FOUT, N] = 64*16*4096 floats

    float* z1 = (float*)d_ws;                        // 64 MB
    float* z2 = z1 + (size_t)NN * NN;                // 64 MB

    const dim3 gGrid(NN / BN, NN / BM, 1);
    const dim3 gBlk(256, 1, 1);
    const int  rGrid = (64 * 16 * (NN / 4)) / 256;   // 4096 blocks

    // k = 0: y = 0.5 * group-reduce(w0 * x)   (initializes y, no memset needed)
    weighted_group_reduce<<<rGrid, gBlk, 0, stream>>>(x, W + 0 * 64, y, 0);

    // z1 = x @ S ; accumulate k = 1
    gemm4096_wmma_f32<<<gGrid, gBlk, 0, stream>>>(x, S, z1);
    weighted_group_reduce<<<rGrid, gBlk, 0, stream>>>(z1, W + 1 * 64, y, 1);

    // z2 = z1 @ S ; accumulate k = 2
    gemm4096_wmma_f32<<<gGrid, gBlk, 0, stream>>>(z1, S, z2);
    weighted_group_reduce<<<rGrid, gBlk, 0, stream>>>(z2, W + 2 * 64, y, 1);

    // z1 <- z2 @ S ; accumulate k = 3
    gemm4096_wmma_f32<<<gGrid, gBlk, 0, stream>>>(z2, S, z1);
    weighted_group_reduce<<<rGrid, gBlk, 0, stream>>>(z1, W + 3 * 64, y, 1);
}